// KVCache_80281528697007
// MI455X (gfx1250) — compile-verified
//
#include <hip/hip_runtime.h>
#include <hip/hip_bf16.h>
#include <cstdint>

// Paged KV-cache append for MI455X (gfx1250, wave32).
// Pure data movement: minimize HBM bytes (1.0 GB instead of naive 1.5 GB)
// and use CDNA5 async global<->LDS DMA (ASYNCcnt) for the bulk copy path.

typedef float __attribute__((ext_vector_type(4))) f4;

// ---------------- CDNA5 async global<->LDS helpers (inline asm) -------------
__device__ __forceinline__ void async_load_b128_to_lds(uint32_t lds_addr, const void* gptr) {
    // global_load_async_to_lds_b128 vdst(lds-addr), vaddr(64b global), off
    asm volatile("global_load_async_to_lds_b128 %0, %1, off"
                 :: "v"(lds_addr), "v"((uint64_t)(uintptr_t)gptr)
                 : "memory");
}
__device__ __forceinline__ void async_store_b128_from_lds(void* gptr, uint32_t lds_addr) {
    // global_store_async_from_lds_b128 vaddr(64b global), vsrc(lds-addr), off
    asm volatile("global_store_async_from_lds_b128 %0, %1, off"
                 :: "v"((uint64_t)(uintptr_t)gptr), "v"(lds_addr)
                 : "memory");
}
__device__ __forceinline__ void wait_asynccnt0() {
    asm volatile("s_wait_asynccnt 0" ::: "memory");
}

// ---------------- Kernel 1: zero the per-(page,slot) coverage flags ---------
__global__ void clear_flags_kernel(unsigned int* __restrict__ w, int nwords) {
    int i = blockIdx.x * blockDim.x + threadIdx.x;
    if (i < nwords) w[i] = 0u;
}

// ---------------- Kernel 2: scatter new K/V tokens into the cache -----------
// One 256-thread block (8 wave32s) per token. Each token's K row and V row are
// 1024 contiguous floats (4 KB) in both src and dst -> each lane moves one
// 16-byte chunk of K and one of V with non-temporal b128 ops.
__global__ void kv_scatter_kernel(const float* __restrict__ k,
                                  const float* __restrict__ v,
                                  const int* __restrict__ app,      // kv_append_indptr [B+1]
                                  const int* __restrict__ pidx,     // kv_page_indices
                                  const int* __restrict__ pip,      // kv_page_indptr  [B+1]
                                  const int* __restrict__ lastlen,  // [B]
                                  const int* __restrict__ psz,      // scalar page_size
                                  float* __restrict__ out,
                                  unsigned char* __restrict__ flags,
                                  int B) {
    const int t = blockIdx.x;
    const int page_size = psz[0];

    // block-uniform ragged search: b with app[b] <= t < app[b+1]
    int b = 0;
    while (b < B - 1 && app[b + 1] <= t) ++b;

    const int num_new   = app[b + 1] - app[b];
    const int num_pages = pip[b + 1] - pip[b];
    const int seq_len   = (num_pages - 1) * page_size + lastlen[b];
    const int pos       = seq_len - num_new + (t - app[b]);
    const int pno       = pos / page_size;
    const int slot      = pos - pno * page_size;
    const int page      = pidx[pip[b] + pno];

    const int HD = 1024;  // H*D = 8*128
    const f4* __restrict__ ks = (const f4*)(k + (size_t)t * HD);
    const f4* __restrict__ vs = (const f4*)(v + (size_t)t * HD);
    f4* kd = (f4*)(out + (((size_t)page * 2 + 0) * page_size + slot) * HD);
    f4* vd = (f4*)(out + (((size_t)page * 2 + 1) * page_size + slot) * HD);

    const int i = threadIdx.x;          // 0..255 == HD/4 chunks
    f4 a = __builtin_nontemporal_load(ks + i);
    f4 c = __builtin_nontemporal_load(vs + i);
    __builtin_nontemporal_store(a, kd + i);
    __builtin_nontemporal_store(c, vd + i);

    if (flags != nullptr && i == 0) {
        flags[(size_t)page * page_size + slot] = (unsigned char)1;
    }
}

// ---------------- Kernel 3: copy untouched (page,slot) rows -----------------
// One block per (page,slot). Covered rows early-exit; untouched rows copy the
// 2 x 4 KB (K row + V row) from the input cache to the output, staged through
// LDS with the CDNA5 async DMA path (per-lane disjoint LDS slots -> no barrier).
__global__ void kv_copy_untouched_kernel(const float* __restrict__ src,
                                         float* __restrict__ dst,
                                         const unsigned char* __restrict__ flags,
                                         const int* __restrict__ psz) {
    const int s = blockIdx.x;                 // page*page_size + slot
    if (flags[s]) return;                     // fully overwritten by scatter

    const int page_size = psz[0];
    const int page = s / page_size;
    const int slot = s - page * page_size;

    const int HD = 1024;
    const size_t koff = (((size_t)page * 2 + 0) * page_size + slot) * HD;
    const size_t voff = (((size_t)page * 2 + 1) * page_size + slot) * HD;

    __shared__ f4 buf[512];                   // 8 KB staging
    const int i = threadIdx.x;                // 0..255

    // Low 32 bits of the flat shared-memory address are the LDS byte offset.
    const uint32_t lk = (uint32_t)(uintptr_t)&buf[i];
    const uint32_t lv = (uint32_t)(uintptr_t)&buf[256 + i];

    const f4* gk = (const f4*)(src + koff) + i;
    const f4* gv = (const f4*)(src + voff) + i;

    async_load_b128_to_lds(lk, gk);
    async_load_b128_to_lds(lv, gv);
    wait_asynccnt0();                         // this wave's LDS data is resident

    async_store_b128_from_lds((f4*)(dst + koff) + i, lk);
    async_store_b128_from_lds((f4*)(dst + voff) + i, lv);
    wait_asynccnt0();                         // drain before LDS/kernel teardown
}

// ---------------------------------------------------------------------------
extern "C" void kernel_launch(void* const* d_in, const int* in_sizes, int n_in,
                              void* d_out, int out_size, void* d_ws, size_t ws_size,
                              hipStream_t stream) {
    const float* k        = (const float*)d_in[0];
    const float* v        = (const float*)d_in[1];
    const float* cache_in = (const float*)d_in[2];
    const int*   app      = (const int*)d_in[3];
    const int*   pidx     = (const int*)d_in[4];
    const int*   pip      = (const int*)d_in[5];
    const int*   lastlen  = (const int*)d_in[6];
    const int*   psz      = (const int*)d_in[7];
    float*       out      = (float*)d_out;

    const int HD   = 1024;              // H*D for this problem instance
    const int PAGE = 16;                // page_size for this problem instance
    const int B    = in_sizes[6];
    const int T    = in_sizes[0] / HD;  // 32768 tokens
    const long cache_elems = (long)in_sizes[2];
    const int NPAGES = (int)(cache_elems / (2L * PAGE * HD));
    const int nslots = NPAGES * PAGE;   // 65536 (page,slot) rows

    unsigned char* flags =
        (ws_size >= (size_t)nslots) ? (unsigned char*)d_ws : nullptr;

    if (flags != nullptr) {
        // Optimized path: scatter + copy only untouched rows (~1.0 GB traffic).
        const int nwords = nslots / 4;
        clear_flags_kernel<<<(nwords + 255) / 256, 256, 0, stream>>>(
            (unsigned int*)flags, nwords);
        kv_scatter_kernel<<<T, 256, 0, stream>>>(
            k, v, app, pidx, pip, lastlen, psz, out, flags, B);
        kv_copy_untouched_kernel<<<nslots, 256, 0, stream>>>(
            cache_in, out, flags, psz);
    } else {
        // Fallback: full d2d copy then scatter (~1.5 GB traffic).
        hipMemcpyAsync(out, cache_in, (size_t)cache_elems * sizeof(float),
                       hipMemcpyDeviceToDevice, stream);
        kv_scatter_kernel<<<T, 256, 0, stream>>>(
            k, v, app, pidx, pip, lastlen, psz, out, nullptr, B);
    }
}